// GCF_FPGA_68032281969033
// MI455X (gfx1250) — compile-verified
//
#include <hip/hip_runtime.h>

typedef float v2f __attribute__((ext_vector_type(2)));
typedef float v8f __attribute__((ext_vector_type(8)));

#define C_CH   512
#define R_CH   128
#define BATCH  8
#define HW     16384          // 128*128
#define PLANES 4096           // BATCH * C_CH
#define BN_EPS 1e-5f

__device__ __forceinline__ float relu6f(float v) {
    return fminf(fmaxf(v, 0.0f), 6.0f);
}

// ---------------------------------------------------------------------------
// K1: per-(b,c) plane mean.  One 256-thread block per 16384-float plane.
// ---------------------------------------------------------------------------
__global__ void __launch_bounds__(256)
se_mean_kernel(const float* __restrict__ x, float* __restrict__ means) {
    const int plane = blockIdx.x;                 // b*C + c
    const float4* xp = (const float4*)(x + (size_t)plane * HW);
    const int t = threadIdx.x;

    float s = 0.0f;
    #pragma unroll
    for (int i = 0; i < 16; ++i) {                // 16 * 256 * 4 = 16384 floats
        float4 v = xp[t + 256 * i];
        s += (v.x + v.y) + (v.z + v.w);
    }
    // wave32 reduction
    #pragma unroll
    for (int off = 16; off > 0; off >>= 1)
        s += __shfl_down(s, off, 32);

    __shared__ float partial[8];
    const int wid = t >> 5, lane = t & 31;
    if (lane == 0) partial[wid] = s;
    __syncthreads();
    if (t == 0) {
        float tot = 0.0f;
        #pragma unroll
        for (int i = 0; i < 8; ++i) tot += partial[i];
        means[plane] = tot * (1.0f / (float)HW);
    }
}

// ---------------------------------------------------------------------------
// K2: fused SE chain on one block (256 threads = 8 wave32).
//   G = relu6( means[16pad x 512] * w_guide^T )        -> [16 x 128]
//   F = G * w_fuse^T                                    -> [16 x 512]
//   bias = relu6( (F - mu) * rsqrt(var+eps)*gamma + beta )
// Uses V_WMMA_F32_16X16X4_F32 (fp32 exact, matches reference precision).
// ---------------------------------------------------------------------------
__global__ void __launch_bounds__(256)
se_excite_kernel(const float* __restrict__ means,
                 const float* __restrict__ w_guide,   // [R, C] row-major
                 const float* __restrict__ w_fuse,    // [C, R] row-major
                 const float* __restrict__ gamma,
                 const float* __restrict__ beta,
                 const float* __restrict__ mu,
                 const float* __restrict__ var,
                 float* __restrict__ bias) {
    __shared__ float Alds[16 * C_CH];   // 32 KB, rows 8..15 zero padded
    __shared__ float Glds[16 * R_CH];   // 8 KB

    const int tid = threadIdx.x;

    // Stage pooled activations, zero-pad batch rows 8..15.
    for (int i = tid; i < 16 * C_CH; i += 256) {
        const int b = i / C_CH;
        Alds[i] = (b < BATCH) ? means[i] : 0.0f;
    }
    __syncthreads();

    const int wid   = tid >> 5;
    const int lane  = tid & 31;
    const int MN    = lane & 15;          // row (A) / col (B,D) within tile
    const int khalf = (lane >> 4) * 2;    // K sub-offset for A/B fragments
    const int rbase = (lane >> 4) * 8;    // D-matrix row base

    // ---- Phase A: guide matmul [16 x 512] x [512 x 128] -> [16 x 128] ----
    {
        const int n0 = wid * 16;          // 8 waves cover N = 128
        v8f acc = {};
        #pragma unroll 4
        for (int k = 0; k < C_CH; k += 4) {
            const int kk = k + khalf;
            v2f a = *(const v2f*)&Alds[MN * C_CH + kk];
            // B[k][n] = w_guide[n][k] -> contiguous float2 along k
            v2f b = *(const v2f*)&w_guide[(size_t)(n0 + MN) * C_CH + kk];
            acc = __builtin_amdgcn_wmma_f32_16x16x4_f32(
                false, a, false, b, (short)0, acc, false, false);
        }
        #pragma unroll
        for (int v = 0; v < 8; ++v)
            Glds[(rbase + v) * R_CH + n0 + MN] = relu6f(acc[v]);
    }
    __syncthreads();

    // ---- Phase B: fuse matmul [16 x 128] x [128 x 512] + BN + relu6 ----
    #pragma unroll
    for (int t4 = 0; t4 < 4; ++t4) {      // 32 N-tiles / 8 waves = 4 each
        const int n0 = (wid + t4 * 8) * 16;
        v8f acc = {};
        #pragma unroll 4
        for (int k = 0; k < R_CH; k += 4) {
            const int kk = k + khalf;
            v2f a = *(const v2f*)&Glds[MN * R_CH + kk];
            v2f b = *(const v2f*)&w_fuse[(size_t)(n0 + MN) * R_CH + kk];
            acc = __builtin_amdgcn_wmma_f32_16x16x4_f32(
                false, a, false, b, (short)0, acc, false, false);
        }
        const int c = n0 + MN;
        const float sc = rsqrtf(var[c] + BN_EPS) * gamma[c];
        const float bb = beta[c] - mu[c] * sc;
        #pragma unroll
        for (int v = 0; v < 8; ++v) {
            const int brow = rbase + v;
            if (brow < BATCH)
                bias[brow * C_CH + c] = relu6f(acc[v] * sc + bb);
        }
    }
}

// ---------------------------------------------------------------------------
// K3: out[b,c,h,w] = x[b,c,h,w] + bias[b,c].  One block per plane, float4.
// ---------------------------------------------------------------------------
__global__ void __launch_bounds__(256)
se_add_kernel(const float* __restrict__ x, const float* __restrict__ bias,
              float* __restrict__ out) {
    const int plane = blockIdx.x;
    const float bv = bias[plane];
    const float4* xp = (const float4*)(x + (size_t)plane * HW);
    float4* op = (float4*)(out + (size_t)plane * HW);
    const int t = threadIdx.x;
    #pragma unroll
    for (int i = 0; i < 16; ++i) {
        float4 v = xp[t + 256 * i];
        v.x += bv; v.y += bv; v.z += bv; v.w += bv;
        op[t + 256 * i] = v;
    }
}

// ---------------------------------------------------------------------------
extern "C" void kernel_launch(void* const* d_in, const int* in_sizes, int n_in,
                              void* d_out, int out_size, void* d_ws, size_t ws_size,
                              hipStream_t stream) {
    const float* x       = (const float*)d_in[0];
    const float* w_guide = (const float*)d_in[1];
    const float* w_fuse  = (const float*)d_in[2];
    const float* gamma   = (const float*)d_in[3];
    const float* beta    = (const float*)d_in[4];
    const float* mu      = (const float*)d_in[5];
    const float* var     = (const float*)d_in[6];
    float* out = (float*)d_out;

    float* means = (float*)d_ws;          // PLANES floats
    float* bias  = means + PLANES;        // PLANES floats

    se_mean_kernel<<<PLANES, 256, 0, stream>>>(x, means);
    se_excite_kernel<<<1, 256, 0, stream>>>(means, w_guide, w_fuse,
                                            gamma, beta, mu, var, bias);
    se_add_kernel<<<PLANES, 256, 0, stream>>>(x, bias, out);
}